// GIN_57440892616781
// MI455X (gfx1250) — compile-verified
//
#include <hip/hip_runtime.h>
#include <hip/hip_bf16.h>
#include <cstdint>

typedef __attribute__((ext_vector_type(2))) float v2f;
typedef __attribute__((ext_vector_type(8))) float v8f;

#define N_NODES 50000
#define DIM     128
#define N_EDGES 800000

// ---------------------------------------------------------------------------
// Simple vectorized copy: dst = src (float4 granularity)
// ---------------------------------------------------------------------------
__global__ __launch_bounds__(256) void gin_copy_f4(float4* __restrict__ dst,
                                                   const float4* __restrict__ src,
                                                   int n4) {
    int i = blockIdx.x * blockDim.x + threadIdx.x;
    if (i < n4) dst[i] = src[i];
}

// ---------------------------------------------------------------------------
// Edge scatter: agg[row] += ea[e] * h[col].  One edge per wave32:
// each lane owns a contiguous float4 of the 128-wide feature row and issues
// 4 native f32 global atomics (forced via inline asm -> global_atomic_add_f32).
// ---------------------------------------------------------------------------
__global__ __launch_bounds__(256) void gin_scatter(float* __restrict__ agg,
                                                   const float* __restrict__ h,
                                                   const long long* __restrict__ ei,
                                                   const float* __restrict__ ea,
                                                   int E) {
    const int lane = threadIdx.x & 31;
    const int e = blockIdx.x * (blockDim.x >> 5) + (threadIdx.x >> 5);
    if (e >= E) return;

    const long long row = ei[e];
    const long long col = ei[(long long)E + e];
    const float w = ea[e];

    const float4 src = ((const float4*)(h + col * DIM))[lane];
    float* dst = agg + row * DIM + lane * 4;

    float vals[4] = {src.x * w, src.y * w, src.z * w, src.w * w};
#pragma unroll
    for (int i = 0; i < 4; ++i) {
        float* p = dst + i;
        asm volatile("global_atomic_add_f32 %0, %1, off"
                     :
                     : "v"(p), "v"(vals[i])
                     : "memory");
    }
}

// ---------------------------------------------------------------------------
// GEMM via V_WMMA_F32_16X16X4_F32:  out[m][n] = sum_k A[m][k] * W[n][k] + b[n]
// One wave computes one 16x16 tile; K=128 -> 32 WMMA instructions.
// Lane mapping (per CDNA5 ISA 7.12.2):
//   lane = 16*half + r
//   A (16x4, MxK): lane loads A[m0+r][k + 2*half + {0,1}]        (contig v2f)
//   B (4x16, KxN): lane loads W[n0+r][k + 2*half + {0,1}]        (contig v2f)
//   D (16x16):     vgpr v, lane -> out[m0 + v + 8*half][n0 + r]
// NCOLS_OUT masks the store (and clamps W-row reads) so EXEC stays all-1s
// through every WMMA.
// ---------------------------------------------------------------------------
template <int NCOLS_OUT, bool RELU>
__global__ __launch_bounds__(256) void gin_gemm_wmma(float* __restrict__ out,
                                                     const float* __restrict__ A,
                                                     const float* __restrict__ W,
                                                     const float* __restrict__ bias,
                                                     int ntiles_n) {
    const int lane = threadIdx.x & 31;
    const int wave = threadIdx.x >> 5;
    if (wave >= ntiles_n) return;  // wave-uniform

    const int half = lane >> 4;
    const int r    = lane & 15;
    const int m0   = blockIdx.x << 4;
    const int n0   = wave << 4;

    const int wrow = (n0 + r < NCOLS_OUT) ? (n0 + r) : (NCOLS_OUT - 1);  // clamp OOB W rows
    const float* arow = A + (size_t)(m0 + r) * DIM + 2 * half;
    const float* brow = W + (size_t)wrow * DIM + 2 * half;

    v8f c = {0.f, 0.f, 0.f, 0.f, 0.f, 0.f, 0.f, 0.f};
#pragma unroll
    for (int k = 0; k < DIM; k += 4) {
        v2f a = *(const v2f*)(arow + k);
        v2f b = *(const v2f*)(brow + k);
        c = __builtin_amdgcn_wmma_f32_16x16x4_f32(
                /*neg_a=*/false, a, /*neg_b=*/false, b,
                /*c_mod=*/(short)0, c, /*reuse_a=*/false, /*reuse_b=*/false);
    }

#pragma unroll
    for (int v = 0; v < 8; ++v) {
        const int row = m0 + v + 8 * half;
        const int col = n0 + r;
        if (col < NCOLS_OUT) {
            float val = c[v] + bias[col];
            if (RELU) val = fmaxf(val, 0.f);
            out[(size_t)row * NCOLS_OUT + col] = val;
        }
    }
}

// ---------------------------------------------------------------------------
// In-place log_softmax over 40 classes, one thread per node.
// ---------------------------------------------------------------------------
__global__ __launch_bounds__(256) void gin_log_softmax(float* __restrict__ out, int nrows) {
    const int n = blockIdx.x * blockDim.x + threadIdx.x;
    if (n >= nrows) return;
    float* row = out + (size_t)n * 40;
    float mx = row[0];
#pragma unroll
    for (int j = 1; j < 40; ++j) mx = fmaxf(mx, row[j]);
    float s = 0.f;
#pragma unroll
    for (int j = 0; j < 40; ++j) s += __expf(row[j] - mx);
    const float lse = __logf(s) + mx;
#pragma unroll
    for (int j = 0; j < 40; ++j) row[j] -= lse;
}

// ---------------------------------------------------------------------------
extern "C" void kernel_launch(void* const* d_in, const int* in_sizes, int n_in,
                              void* d_out, int out_size, void* d_ws, size_t ws_size,
                              hipStream_t stream) {
    const float* x        = (const float*)d_in[0];
    const long long* ei   = (const long long*)d_in[1];  // int64 edge_index (2, E)
    const float* ea       = (const float*)d_in[2];
    const float* W1       = (const float*)d_in[3];
    const float* b1       = (const float*)d_in[4];
    const float* W2       = (const float*)d_in[5];
    const float* b2       = (const float*)d_in[6];
    float* out            = (float*)d_out;

    float* agg = (float*)d_ws;                  // [N_NODES, DIM]
    float* h   = agg + (size_t)N_NODES * DIM;   // [N_NODES, DIM]

    const int n4 = N_NODES * DIM / 4;
    const int copy_grid    = (n4 + 255) / 256;
    const int scatter_grid = (N_EDGES + 7) / 8;   // 8 waves (edges) per 256-thread block
    const int gemm_grid    = N_NODES / 16;        // 3125 exact

    // ---- Layer 1: agg = x + A@x ; h = relu(agg @ W1^T + b1) ----
    gin_copy_f4<<<copy_grid, 256, 0, stream>>>((float4*)agg, (const float4*)x, n4);
    gin_scatter<<<scatter_grid, 256, 0, stream>>>(agg, x, ei, ea, N_EDGES);
    gin_gemm_wmma<128, true><<<gemm_grid, 256, 0, stream>>>(h, agg, W1, b1, 8);

    // ---- Layer 2: agg = h + A@h ; out = agg @ W2^T + b2 ----
    gin_copy_f4<<<copy_grid, 256, 0, stream>>>((float4*)agg, (const float4*)h, n4);
    gin_scatter<<<scatter_grid, 256, 0, stream>>>(agg, h, ei, ea, N_EDGES);
    gin_gemm_wmma<40, false><<<gemm_grid, 96, 0, stream>>>(out, agg, W2, b2, 3);

    // ---- log_softmax in place on d_out ----
    gin_log_softmax<<<(N_NODES + 255) / 256, 256, 0, stream>>>(out, N_NODES);
}